// BlockDCT_71889162600923
// MI455X (gfx1250) — compile-verified
//
#include <hip/hip_runtime.h>

typedef __attribute__((ext_vector_type(2))) float v2f;
typedef __attribute__((ext_vector_type(4))) float v4f;
typedef __attribute__((ext_vector_type(8))) float v8f;
typedef __attribute__((ext_vector_type(4))) int   v4i;

#define PI_F 3.1415f  // faithful to the reference module's PI

// One workgroup (128 threads = 4 wave32) handles one (bc, bh) pair:
//   - async-copies the contiguous 8x1024 fp32 image strip into LDS (32 KB)
//   - builds the 64x64 fp32 DCT operator W in LDS (rows padded to 66)
//   - computes Y[64,128] = W[64,64] x X[64,128] with v_wmma_f32_16x16x4_f32
//   - stages the 64x128 result tile back through LDS for b128-coalesced stores
// Wave w owns M-tile rows [16w,16w+16); K=64 as 16 steps of K=4; 8 N-subtiles.
__global__ __launch_bounds__(128) void dct8_wmma_kernel(const float* __restrict__ x,
                                                        float* __restrict__ out) {
  __shared__ float Wlds[64 * 66];   // 64x64 operator, padded stride 66
  __shared__ float patch[8 * 1024]; // in: 8x1024 strip; later reused: out stage 64x128

  const int bh  = blockIdx.x;   // 0..127 block-row
  const int bc  = blockIdx.y;   // 0..95  (b*3 + c)
  const int tid = threadIdx.x;  // 0..127

  // ---- kick off async global->LDS DMA of the 8x1024 strip (fully contiguous)
  const float* src = x + (size_t)bc * (1024u * 1024u) + (size_t)bh * 8192u;
#if __has_builtin(__builtin_amdgcn_global_load_async_to_lds_b128)
  for (int i = tid; i < 2048; i += 128) {
    __builtin_amdgcn_global_load_async_to_lds_b128(
        (__attribute__((address_space(1))) v4i*)(src + 4 * i),
        (__attribute__((address_space(3))) v4i*)(patch + 4 * i),
        0, 0);
  }
#endif

  // ---- build W[row=u*8+v][col=r*8+s] = cos((2u+1)r*pi/16)*cos((2v+1)s*pi/16)
  // (overlaps with the async DMA above)
  for (int idx = tid; idx < 64 * 64; idx += 128) {
    const int row = idx >> 6, col = idx & 63;
    const int u = row >> 3, v = row & 7;
    const int r = col >> 3, s = col & 7;
    const float au = cosf((2.0f * u + 1.0f) * (PI_F / 16.0f) * (float)r);
    const float av = cosf((2.0f * v + 1.0f) * (PI_F / 16.0f) * (float)s);
    Wlds[row * 66 + col] = au * av;
  }

#if __has_builtin(__builtin_amdgcn_global_load_async_to_lds_b128)
#if __has_builtin(__builtin_amdgcn_s_wait_asynccnt)
  __builtin_amdgcn_s_wait_asynccnt(0);
#else
  asm volatile("s_wait_asynccnt 0" ::: "memory");
#endif
#else
  // fallback: plain float4 copy through VGPRs
  for (int i = tid; i < 2048; i += 128) {
    ((v4f*)patch)[i] = ((const v4f*)src)[i];
  }
#endif
  __syncthreads();

  const int wave = tid >> 5;
  const int lane = tid & 31;
  const int half = lane >> 4;   // which 16-lane half of the wave
  const int lm   = lane & 15;

  // ---- preload all 16 A fragments for this wave's M-tile
  // A layout (16x4 f32): lanes 0-15 M=lm, comps = K{0,1} of window;
  //                      lanes 16-31 M=lm, comps = K{2,3}  -> k0 = 4kk + 2*half
  v2f afrag[16];
#pragma unroll
  for (int kk = 0; kk < 16; ++kk) {
    const int k0 = 4 * kk + 2 * half;
    const float* wr = &Wlds[(16 * wave + lm) * 66 + k0];
    afrag[kk].x = wr[0];
    afrag[kk].y = wr[1];
  }

  // ---- 8 N-subtiles of 16 block-columns each; keep all accumulators live
  v8f acc[8];
#pragma unroll
  for (int st = 0; st < 8; ++st) {
    const int n = st * 16 + lm;  // global block-column (bw) for this lane
    acc[st] = (v8f){};
#pragma unroll
    for (int kk = 0; kk < 16; ++kk) {
      const int k0 = 4 * kk + 2 * half;       // even -> (s, s+1) same row r
      const int r = k0 >> 3, s = k0 & 7;
      const float* bp = &patch[r * 1024 + n * 8 + s];
      v2f bfrag;
      bfrag.x = bp[0];
      bfrag.y = bp[1];
      // D = A(16x4) * B(4x16) + C, fp32 end-to-end
      acc[st] = __builtin_amdgcn_wmma_f32_16x16x4_f32(
          false, afrag[kk], false, bfrag, (short)0, acc[st], false, false);
    }
  }

  // ---- stage the 64x128 tile through LDS (reuse patch) for coalesced stores
  __syncthreads();  // all waves done reading B fragments from patch
  float* stage = patch;
#pragma unroll
  for (int st = 0; st < 8; ++st) {
#pragma unroll
    for (int i = 0; i < 8; ++i) {
      // C/D layout: VGPR i -> M = i (lanes 0-15) / i+8 (lanes 16-31), N = lm
      const int row = 16 * wave + 8 * half + i;
      stage[row * 128 + st * 16 + lm] = acc[st][i];
    }
  }
  __syncthreads();

  // ---- b128 stores: row r of stage -> out[((bc*64+r)*128 + bh)*128 + 0..127]
  const size_t obase = (size_t)bc * 64 * (128 * 128) + (size_t)bh * 128;
  for (int j = tid; j < 2048; j += 128) {
    const int row = j >> 5;   // 0..63
    const int cg  = j & 31;   // float4 group within the 128-col row
    const v4f val = ((const v4f*)stage)[j];
    *(v4f*)(out + obase + (size_t)row * (128 * 128) + cg * 4) = val;
  }
}

extern "C" void kernel_launch(void* const* d_in, const int* in_sizes, int n_in,
                              void* d_out, int out_size, void* d_ws, size_t ws_size,
                              hipStream_t stream) {
  (void)in_sizes; (void)n_in; (void)out_size; (void)d_ws; (void)ws_size;
  const float* x = (const float*)d_in[0];
  float* out = (float*)d_out;
  dim3 grid(128, 96);   // bh x (b*3+c)
  dim3 block(128);      // 4 wave32
  dct8_wmma_kernel<<<grid, block, 0, stream>>>(x, out);
}